// GCNEncoderLayer_17171279249942
// MI455X (gfx1250) — compile-verified
//
#include <hip/hip_runtime.h>
#include <hip/hip_bf16.h>
#include <stdint.h>

typedef __attribute__((ext_vector_type(2))) float v2f;
typedef __attribute__((ext_vector_type(8))) float v8f;

#define D 128

// ---------------------------------------------------------------------------
// 1) degree init: deg[i] = 1.0 (self loop)
__global__ void gcn_deg_init(float* __restrict__ deg, int N) {
  int i = blockIdx.x * blockDim.x + threadIdx.x;
  if (i < N) deg[i] = 1.0f;
}

// 2) degree count: deg[col[e]] += 1
__global__ void gcn_deg_count(const long long* __restrict__ ei, float* __restrict__ deg, int E) {
  int e = blockIdx.x * blockDim.x + threadIdx.x;
  if (e < E) {
    int col = (int)ei[(size_t)E + e];
    unsafeAtomicAdd(&deg[col], 1.0f);
  }
}

// 3) dinv[i] = rsqrt(deg[i])   (deg >= 1 always, self loop included)
__global__ void gcn_deg_to_dinv(float* __restrict__ deg, int N) {
  int i = blockIdx.x * blockDim.x + threadIdx.x;
  if (i < N) deg[i] = __frsqrt_rn(deg[i]);
}

// ---------------------------------------------------------------------------
// 4) h = x @ W via V_WMMA_F32_16X16X4_F32.
//    Block = 256 threads = 8 wave32; each wave owns a 16-row strip, all 8
//    column tiles (8 x v8f accumulators).
//    W staged in LDS as k-pair float2s with a PARITY-XOR swizzle:
//      Wp[kp][n] stored at column  n ^ ((kp & 1) << 5).
//    The two half-waves read rows kp (even) and kp+1 (odd), so each lane's
//    swizzle is compile-time constant and the odd half's 16-wide bank range
//    is XOR-32'd into the disjoint complement -> conflict-free ds_load_b64.
//    Full k unroll: all DS/global addresses fold into offset immediates.
__global__ __launch_bounds__(256) void gcn_gemm_wmma(const float* __restrict__ x,
                                                     const float* __restrict__ W,
                                                     float* __restrict__ h, int N) {
  __shared__ float Wlds[D * D];  // 64 kpairs x 128 float2 = 64KB

  // cooperative paired+swizzled store of W into LDS
  for (int idx = threadIdx.x; idx < D * D; idx += 256) {
    int k = idx >> 7;
    int n = idx & 127;
    int kp = k >> 1;
    int ecol = n ^ ((kp & 1) << 5);
    Wlds[(((kp << 7) | ecol) << 1) + (k & 1)] = W[idx];
  }
  __syncthreads();

  const v2f* __restrict__ Wp = (const v2f*)Wlds;

  const int lane  = threadIdx.x & 31;
  const int wave  = threadIdx.x >> 5;
  const int r     = lane & 15;   // A row-in-tile; B/C/D col-in-tile
  const int khalf = lane >> 4;   // 0: K pair {0,1}; 1: K pair {2,3}
  const int mbase = (blockIdx.x * 8 + wave) * 16;
  int row = mbase + r;
  if (row >= N) row = N - 1;     // clamp: OOB lanes feed rows that are never stored
  const float* __restrict__ xa = x + (size_t)row * D + 2 * khalf;  // A base, offsets k*16B

  // 8 per-tile LDS bases: element index = khalf*128 + ((t ^ 2*khalf)<<4) + r,
  // plus k<<6 elements (= k*512 bytes) per k-step, folded into DS offsets.
  const v2f* wb[8];
#pragma unroll
  for (int t = 0; t < 8; ++t) {
    wb[t] = Wp + (khalf << 7) + (((t ^ (khalf << 1)) << 4) + r);
  }

  v8f acc[8];
  v8f zero = {};
#pragma unroll
  for (int t = 0; t < 8; ++t) acc[t] = zero;

#pragma unroll
  for (int k = 0; k < D; k += 4) {
    const v2f a = *(const v2f*)(xa + k);
    v2f bf[8];
#pragma unroll
    for (int t = 0; t < 8; ++t) bf[t] = wb[t][k << 6];
#pragma unroll
    for (int t = 0; t < 8; ++t) {
      acc[t] = __builtin_amdgcn_wmma_f32_16x16x4_f32(
          /*neg_a=*/false, a, /*neg_b=*/false, bf[t],
          /*c_mod=*/(short)0, acc[t], /*reuse_a=*/false, /*reuse_b=*/false);
    }
  }

  // D layout: VGPR v -> M = mbase + v + 8*khalf, N = r
  const int m0 = mbase + 8 * khalf;
#pragma unroll
  for (int v = 0; v < 8; ++v) {
    const int m = m0 + v;
    if (m < N) {
#pragma unroll
      for (int t = 0; t < 8; ++t) {
        h[(size_t)m * D + t * 16 + r] = acc[t][v];
      }
    }
  }
}

// ---------------------------------------------------------------------------
// 5) out[i][d] = h[i][d] * dinv[i]^2   (self-loop message; overwrites d_out)
//    float4-vectorized: 32 float4 per row.
__global__ void gcn_self_loop(const float4* __restrict__ h4, const float* __restrict__ dinv,
                              float4* __restrict__ out4, int total4) {
  int g = blockIdx.x * blockDim.x + threadIdx.x;
  if (g < total4) {
    int i = g >> 5;
    float di = dinv[i];
    float s = di * di;
    float4 v = h4[g];
    out4[g] = make_float4(v.x * s, v.y * s, v.z * s, v.w * s);
  }
}

// 6) scatter-add edge messages: one thread per (edge, feature).
//    128 consecutive threads = one edge: contiguous 512B h read + contiguous
//    global_atomic_add_f32 burst into out[col].
__global__ void gcn_scatter(const float* __restrict__ h, const long long* __restrict__ ei,
                            const float* __restrict__ dinv, float* __restrict__ out, int E) {
  long long g = (long long)blockIdx.x * blockDim.x + threadIdx.x;
  long long total = (long long)E << 7;
  if (g >= total) return;
  int e = (int)(g >> 7);
  int d = (int)(g & 127);
  int row = (int)ei[e];
  int col = (int)ei[(size_t)E + e];
  float norm = dinv[row] * dinv[col];
  unsafeAtomicAdd(&out[(size_t)col * D + d], h[(size_t)row * D + d] * norm);
}

// 7) out = relu(out + b), in place, float4-vectorized
__global__ void gcn_bias_relu(float4* __restrict__ out4, const float4* __restrict__ b4,
                              int total4) {
  int g = blockIdx.x * blockDim.x + threadIdx.x;
  if (g < total4) {
    float4 v = out4[g];
    float4 bb = b4[g & 31];
    v.x = fmaxf(v.x + bb.x, 0.0f);
    v.y = fmaxf(v.y + bb.y, 0.0f);
    v.z = fmaxf(v.z + bb.z, 0.0f);
    v.w = fmaxf(v.w + bb.w, 0.0f);
    out4[g] = v;
  }
}

// ---------------------------------------------------------------------------
extern "C" void kernel_launch(void* const* d_in, const int* in_sizes, int n_in,
                              void* d_out, int out_size, void* d_ws, size_t ws_size,
                              hipStream_t stream) {
  const float*     x  = (const float*)d_in[0];
  const long long* ei = (const long long*)d_in[1];
  const float*     W  = (const float*)d_in[2];
  const float*     b  = (const float*)d_in[3];
  float* out = (float*)d_out;

  const int N = in_sizes[0] / D;  // 50000
  const int E = in_sizes[1] / 2;  // 800000

  // workspace layout: dinv [N floats] | h [N*D floats]
  char* ws = (char*)d_ws;
  float* dinv = (float*)ws;
  size_t off = (((size_t)N * sizeof(float)) + 255) & ~(size_t)255;
  float* h = (float*)(ws + off);

  const int B256 = 256;
  const int total4 = N * (D / 4);  // float4 count of out/h

  gcn_deg_init<<<(N + B256 - 1) / B256, B256, 0, stream>>>(dinv, N);
  gcn_deg_count<<<(E + B256 - 1) / B256, B256, 0, stream>>>(ei, dinv, E);
  gcn_deg_to_dinv<<<(N + B256 - 1) / B256, B256, 0, stream>>>(dinv, N);

  // 8 waves * 16 rows = 128 rows per block
  gcn_gemm_wmma<<<(N + 127) / 128, 256, 0, stream>>>(x, W, h, N);

  gcn_self_loop<<<(total4 + B256 - 1) / B256, B256, 0, stream>>>(
      (const float4*)h, dinv, (float4*)out, total4);

  long long totalED = (long long)E * D;
  gcn_scatter<<<(unsigned)((totalED + B256 - 1) / B256), B256, 0, stream>>>(h, ei, dinv, out, E);

  gcn_bias_relu<<<(total4 + B256 - 1) / B256, B256, 0, stream>>>(
      (float4*)out, (const float4*)b, total4);
}